// AsyncTFCriterion_62225486185080
// MI455X (gfx1250) — compile-verified
//
#include <hip/hip_runtime.h>

namespace {

constexpr int Bz = 256;
constexpr int NS = 160;
constexpr int NO = 320;
constexpr int NV = 480;
constexpr int Mh = 20;
constexpr float W_T = 0.1f;
constexpr float W_S = 0.1f;
constexpr float INV2SIG2 = 1.0f / (2.0f * 300.0f * 300.0f);

typedef __attribute__((ext_vector_type(2))) float v2f;
typedef __attribute__((ext_vector_type(8))) float v8f;

__device__ __forceinline__ void gAdd(float* p, float v) {
  __hip_atomic_fetch_add(p, v, __ATOMIC_RELAXED, __HIP_MEMORY_SCOPE_AGENT);
}
__device__ __forceinline__ void sAdd(float* p, float v) {
  __hip_atomic_fetch_add(p, v, __ATOMIC_RELAXED, __HIP_MEMORY_SCOPE_WORKGROUP);
}

// ---------------- Kernel A: Gaussian temporal messages (past + future fused) ----------------
__global__ void msg_kernel(const int* __restrict__ ids, const int* __restrict__ times,
                           const float* __restrict__ mem_s, const float* __restrict__ mem_o,
                           const float* __restrict__ mem_v, const int* __restrict__ mem_t,
                           const unsigned char* __restrict__ mem_valid,
                           float* __restrict__ s_msg, float* __restrict__ o_msg, float* __restrict__ v_msg,
                           float* __restrict__ s_f, float* __restrict__ o_f, float* __restrict__ v_f) {
  int b = blockIdx.x;
  int tid = threadIdx.x;
  int id = ids[b];
  float t0 = (float)times[b];
  __shared__ float wp[Mh], wf[Mh];
  __shared__ int cntp, cntf;
  __shared__ float invp, invf;
  if (tid == 0) { cntp = 0; cntf = 0; }
  __syncthreads();
  if (tid < Mh) {
    float th = (float)mem_t[id * Mh + tid];
    bool valid = mem_valid[id * Mh + tid] != 0;
    float dt = th - t0;
    float g = expf(-dt * dt * INV2SIG2);
    bool mp = valid && (th < t0);
    bool mf = valid && (th > t0);
    wp[tid] = mp ? g : 0.0f;
    wf[tid] = mf ? g : 0.0f;
    if (mp) atomicAdd(&cntp, 1);
    if (mf) atomicAdd(&cntf, 1);
  }
  __syncthreads();
  if (tid == 0) {
    invp = (cntp > 0) ? 1.0f / (float)cntp : 0.0f;
    invf = (cntf > 0) ? 1.0f / (float)cntf : 0.0f;
  }
  __syncthreads();
  const float* ms = mem_s + (size_t)id * Mh * NS;
  for (int c = tid; c < NS; c += blockDim.x) {
    float sp = 0.0f, sf = 0.0f;
#pragma unroll
    for (int m = 0; m < Mh; ++m) { float val = ms[m * NS + c]; sp += wp[m] * val; sf += wf[m] * val; }
    s_msg[b * NS + c] = sp * invp;
    s_f[b * NS + c] = sf * invf;
  }
  const float* mo = mem_o + (size_t)id * Mh * NO;
  for (int c = tid; c < NO; c += blockDim.x) {
    float sp = 0.0f, sf = 0.0f;
#pragma unroll
    for (int m = 0; m < Mh; ++m) { float val = mo[m * NO + c]; sp += wp[m] * val; sf += wf[m] * val; }
    o_msg[b * NO + c] = sp * invp;
    o_f[b * NO + c] = sf * invf;
  }
  const float* mv = mem_v + (size_t)id * Mh * NV;
  for (int c = tid; c < NV; c += blockDim.x) {
    float sp = 0.0f, sf = 0.0f;
#pragma unroll
    for (int m = 0; m < Mh; ++m) { float val = mv[m * NV + c]; sp += wp[m] * val; sf += wf[m] * val; }
    v_msg[b * NV + c] = sp * invp;
    v_f[b * NV + c] = sf * invf;
  }
}

// ---------------- Softmax + cross-entropy (NS=160, one block per sample) ----------------
__global__ void softmax_ce_kernel(const float* __restrict__ base, const float* __restrict__ acc,
                                  const int* __restrict__ tgt, float* __restrict__ qout,
                                  float* __restrict__ loss, float scale) {
  int b = blockIdx.x, tid = threadIdx.x;
  float t = base[b * NS + tid];
  if (acc) t += acc[b * NS + tid];
  __shared__ float red[NS];
  red[tid] = t;
  __syncthreads();
  for (int s = 128; s > 0; s >>= 1) {
    if (tid < s && tid + s < NS) red[tid] = fmaxf(red[tid], red[tid + s]);
    __syncthreads();
  }
  float mx = red[0];
  __syncthreads();
  float e = expf(t - mx);
  red[tid] = e;
  __syncthreads();
  for (int s = 128; s > 0; s >>= 1) {
    if (tid < s && tid + s < NS) red[tid] += red[tid + s];
    __syncthreads();
  }
  float sum = red[0];
  qout[b * NS + tid] = e / sum;
  if (tid == tgt[b]) gAdd(loss, scale * (mx + logf(sum) - t));
}

// ---------------- Sigmoid + BCE-with-logits (one block per sample, N threads) ----------------
__global__ void sigmoid_bce_kernel(const float* __restrict__ base, const float* __restrict__ acc,
                                   const float* __restrict__ tgt, float* __restrict__ qout,
                                   float* __restrict__ loss, int N, float scale) {
  int b = blockIdx.x, c = threadIdx.x;
  float z = base[b * N + c];
  if (acc) z += acc[b * N + c];
  qout[b * N + c] = 1.0f / (1.0f + expf(-z));
  float bce = fmaxf(z, 0.0f) - z * tgt[b * N + c] + log1pf(expf(-fabsf(z)));
  __shared__ float red[NV];
  red[c] = bce;
  __syncthreads();
  for (int s = 256; s > 0; s >>= 1) {
    if (c < s && c + s < N) red[c] += red[c + s];
    __syncthreads();
  }
  if (c == 0) gAdd(loss, scale * red[0]);
}

// ---------------- Kernel C: single-pass stream, colsum + rowsum both via WMMA ----------------
// colsum[c] = sum_r a[r]*M[r,c]  -> tile as B-operand (rows in K), A = a-chunk replicated over M
// rowsum[r] = sum_c M[r,c]*x[c]  -> tile as A-operand (cols in K), B = x-chunk replicated over N
struct MatDesc {
  const float* mat;
  const float* colvec;  // len R
  const float* rowvec;  // len C
  float* colacc;        // len C target accumulator
  float* rowacc;        // len R target accumulator
  int R;
  int C;
  int shape;            // compile-time (R,C) dispatch id
  float wcol;
  float wrow;
};
struct Tile { int m; int cbase; };
struct CParams { MatDesc md[12]; Tile tiles[24]; };

template <int R, int C>
__device__ __forceinline__ void fused_body(const float* __restrict__ Mb,
                                           const float* __restrict__ a,
                                           const float* __restrict__ x,
                                           float* __restrict__ colacc,
                                           float* __restrict__ rowacc,
                                           float wcol, float wrow, int cbase,
                                           float* lds_a, float* lds_x, float* lds_row) {
  for (int i = threadIdx.x; i < R; i += 320) { lds_a[i] = a[i]; lds_row[i] = 0.0f; }
  for (int i = threadIdx.x; i < 160; i += 320) lds_x[i] = x[cbase + i];
  __syncthreads();

  const int lane = threadIdx.x & 31;
  const int wave = threadIdx.x >> 5;   // 0..9 -> one 16-column strip each
  const int l15 = lane & 15;
  const int kh2 = (lane >> 4) << 1;    // 0 (lanes 0-15) or 2 (lanes 16-31)
  const int c0 = cbase + wave * 16;    // column base of this wave's strip
  const int cl = wave * 16;            // local column base into lds_x

  // Hoisted rowsum B-operands: B2_j[k,n] = x[c0+4j+k] (k pairs per lane-half), loop-invariant.
  v2f xb[4];
#pragma unroll
  for (int j = 0; j < 4; ++j) {
    xb[j].x = lds_x[cl + 4 * j + kh2];
    xb[j].y = lds_x[cl + 4 * j + kh2 + 1];
  }

  v8f dacc = {};  // colsum accumulator: D[m,n] = colsum(c0+n), identical over m
  // B-layout stream pointer: row (rr + 4j + kh2), col c0+l15   (offsets now immediates)
  const float* bp = Mb + (size_t)kh2 * C + c0 + l15;
  // A-layout stream pointer: row (rr + l15), col c0 + kh2 (+4j)
  const float* ap = Mb + (size_t)l15 * C + c0 + kh2;

  for (int rr = 0; rr < R; rr += 16) {
    v8f eacc = {};  // rowsum accumulator for rows rr..rr+15 of this wave's 16-col strip
#pragma unroll
    for (int j = 0; j < 4; ++j) {
      // ---- colsum substep: rows rr+4j+kh2, rr+4j+kh2+1 ----
      float v0 = bp[4 * j * C];
      float v1 = bp[(4 * j + 1) * C];
      v2f A;  A.x = lds_a[rr + 4 * j + kh2];  A.y = lds_a[rr + 4 * j + kh2 + 1];
      v2f Bv; Bv.x = v0;                      Bv.y = v1;
      dacc = __builtin_amdgcn_wmma_f32_16x16x4_f32(false, A, false, Bv, (short)0, dacc, false, false);
      // ---- rowsum substep: same tile re-read in A layout (L0/L2 hit) ----
      float2 av = *reinterpret_cast<const float2*>(ap + 4 * j);
      v2f A2; A2.x = av.x; A2.y = av.y;
      eacc = __builtin_amdgcn_wmma_f32_16x16x4_f32(false, A2, false, xb[j], (short)0, eacc, false, false);
    }
    bp += 16 * C;
    ap += 16 * C;
    // Flush rowsums: lane 0 holds rows rr..rr+7 in comps 0..7, lane 16 holds rr+8..rr+15.
    if (l15 == 0) {
      int rbase = rr + (kh2 << 2);  // kh2=0 -> rr, kh2=2 -> rr+8
      sAdd(&lds_row[rbase + 0], eacc[0]);
      sAdd(&lds_row[rbase + 1], eacc[1]);
      sAdd(&lds_row[rbase + 2], eacc[2]);
      sAdd(&lds_row[rbase + 3], eacc[3]);
      sAdd(&lds_row[rbase + 4], eacc[4]);
      sAdd(&lds_row[rbase + 5], eacc[5]);
      sAdd(&lds_row[rbase + 6], eacc[6]);
      sAdd(&lds_row[rbase + 7], eacc[7]);
    }
  }
  // colsum flush: lane n (0..15) holds D[*, n] = colsum(c0+n) in component 0
  if (lane < 16) gAdd(colacc + c0 + lane, wcol * dacc[0]);
  __syncthreads();
  for (int i = threadIdx.x; i < R; i += 320) gAdd(rowacc + i, wrow * lds_row[i]);
}

__global__ __launch_bounds__(320) void fused_matvec_kernel(CParams p) {
  const int b = blockIdx.y;
  const Tile tl = p.tiles[blockIdx.x];
  const MatDesc d = p.md[tl.m];
  __shared__ __align__(16) float lds_a[NV];
  __shared__ __align__(16) float lds_x[160];
  __shared__ float lds_row[NV];
  const float* Mb = d.mat + (size_t)b * d.R * d.C;
  const float* a = d.colvec + (size_t)b * d.R;
  const float* x = d.rowvec + (size_t)b * d.C;
  float* ca = d.colacc + (size_t)b * d.C;
  float* ra = d.rowacc + (size_t)b * d.R;
  switch (d.shape) {  // uniform per block
    case 0: fused_body<160, 160>(Mb, a, x, ca, ra, d.wcol, d.wrow, tl.cbase, lds_a, lds_x, lds_row); break;
    case 1: fused_body<320, 320>(Mb, a, x, ca, ra, d.wcol, d.wrow, tl.cbase, lds_a, lds_x, lds_row); break;
    case 2: fused_body<480, 480>(Mb, a, x, ca, ra, d.wcol, d.wrow, tl.cbase, lds_a, lds_x, lds_row); break;
    case 3: fused_body<160, 320>(Mb, a, x, ca, ra, d.wcol, d.wrow, tl.cbase, lds_a, lds_x, lds_row); break;
    case 4: fused_body<320, 480>(Mb, a, x, ca, ra, d.wcol, d.wrow, tl.cbase, lds_a, lds_x, lds_row); break;
    case 5: fused_body<480, 160>(Mb, a, x, ca, ra, d.wcol, d.wrow, tl.cbase, lds_a, lds_x, lds_row); break;
    case 6: fused_body<320, 160>(Mb, a, x, ca, ra, d.wcol, d.wrow, tl.cbase, lds_a, lds_x, lds_row); break;
    case 7: fused_body<480, 320>(Mb, a, x, ca, ra, d.wcol, d.wrow, tl.cbase, lds_a, lds_x, lds_row); break;
    case 8: fused_body<160, 480>(Mb, a, x, ca, ra, d.wcol, d.wrow, tl.cbase, lds_a, lds_x, lds_row); break;
    default: break;
  }
}

}  // namespace

extern "C" void kernel_launch(void* const* d_in, const int* in_sizes, int n_in,
                              void* d_out, int out_size, void* d_ws, size_t ws_size,
                              hipStream_t stream) {
  (void)in_sizes; (void)n_in; (void)out_size; (void)ws_size;
  const float* s = (const float*)d_in[0];
  const float* o = (const float*)d_in[1];
  const float* v = (const float*)d_in[2];
  const float* so = (const float*)d_in[3];
  const float* ov = (const float*)d_in[4];
  const float* vs = (const float*)d_in[5];
  const float* ss = (const float*)d_in[6];
  const float* oo = (const float*)d_in[7];
  const float* vv = (const float*)d_in[8];
  const float* so_t = (const float*)d_in[9];
  const float* ov_t = (const float*)d_in[10];
  const float* vs_t = (const float*)d_in[11];
  const float* os_t = (const float*)d_in[12];
  const float* vo_t = (const float*)d_in[13];
  const float* sv_t = (const float*)d_in[14];
  const int* s_target = (const int*)d_in[15];
  const float* o_target = (const float*)d_in[16];
  const float* v_target = (const float*)d_in[17];
  const int* ids = (const int*)d_in[18];
  const int* times = (const int*)d_in[19];
  const float* mem_s = (const float*)d_in[20];
  const float* mem_o = (const float*)d_in[21];
  const float* mem_v = (const float*)d_in[22];
  const int* mem_times = (const int*)d_in[23];
  const unsigned char* mem_valid = (const unsigned char*)d_in[24];

  float* w = (float*)d_ws;
  float* s_msg = w;
  float* o_msg = s_msg + Bz * NS;
  float* v_msg = o_msg + Bz * NO;
  float* s_f = v_msg + Bz * NV;
  float* o_f = s_f + Bz * NS;
  float* v_f = o_f + Bz * NO;
  float* qs_ = v_f + Bz * NV;
  float* qo_ = qs_ + Bz * NS;
  float* qv_ = qo_ + Bz * NO;
  float* accS = qv_ + Bz * NV;
  float* accO = accS + Bz * NS;
  float* accV = accO + Bz * NO;

  float* out = (float*)d_out;
  float* out_qs = out;
  float* out_qo = out + Bz * NS;
  float* out_qv = out_qo + Bz * NO;
  float* out_loss = out_qv + Bz * NV;

  hipMemsetAsync(accS, 0, sizeof(float) * Bz * (NS + NO + NV), stream);
  hipMemsetAsync(out_loss, 0, sizeof(float), stream);

  msg_kernel<<<Bz, 256, 0, stream>>>(ids, times, mem_s, mem_o, mem_v, mem_times, mem_valid,
                                     s_msg, o_msg, v_msg, s_f, o_f, v_f);
  softmax_ce_kernel<<<Bz, NS, 0, stream>>>(s, nullptr, s_target, qs_, out_loss, 1.0f / Bz);
  sigmoid_bce_kernel<<<Bz, NO, 0, stream>>>(o, nullptr, o_target, qo_, out_loss, NO, 1.0f / (Bz * NO));
  sigmoid_bce_kernel<<<Bz, NV, 0, stream>>>(v, nullptr, v_target, qv_, out_loss, NV, 1.0f / (Bz * NV));

  CParams p;
  auto set = [&](int i, const float* mat, int R, int C, int shape, const float* cv, float* ca,
                 float wc, const float* rv, float* ra, float wr) {
    p.md[i] = {mat, cv, rv, ca, ra, R, C, shape, wc, wr};
  };
  //    mat    R   C  shp colvec colacc wcol  rowvec rowacc wrow
  set(0, ss,   NS, NS, 0, s_msg, accS, W_T, s_f, accS, W_T);
  set(1, oo,   NO, NO, 1, o_msg, accO, W_T, o_f, accO, W_T);
  set(2, vv,   NV, NV, 2, v_msg, accV, W_T, v_f, accV, W_T);
  set(3, so,   NS, NO, 3, qs_,  accO, W_S, qo_, accS, W_S);
  set(4, ov,   NO, NV, 4, qo_,  accV, W_S, qv_, accO, W_S);
  set(5, vs,   NV, NS, 5, qv_,  accS, W_S, qs_, accV, W_S);
  set(6, so_t, NS, NO, 3, s_msg, accO, W_T, o_f, accS, W_T);
  set(7, ov_t, NO, NV, 4, o_msg, accV, W_T, v_f, accO, W_T);
  set(8, vs_t, NV, NS, 5, v_msg, accS, W_T, s_f, accV, W_T);
  set(9, os_t, NO, NS, 6, o_msg, accS, W_T, s_f, accO, W_T);
  set(10, vo_t, NV, NO, 7, v_msg, accO, W_T, o_f, accV, W_T);
  set(11, sv_t, NS, NV, 8, s_msg, accV, W_T, v_f, accS, W_T);
  int nt = 0;
  for (int i = 0; i < 12; ++i)
    for (int cb = 0; cb < p.md[i].C; cb += 160) p.tiles[nt++] = {i, cb};
  // nt == 24
  fused_matvec_kernel<<<dim3(24, Bz), 320, 0, stream>>>(p);

  softmax_ce_kernel<<<Bz, NS, 0, stream>>>(s, accS, s_target, out_qs, out_loss, 1.0f / Bz);
  sigmoid_bce_kernel<<<Bz, NO, 0, stream>>>(o, accO, o_target, out_qo, out_loss, NO, 1.0f / (Bz * NO));
  sigmoid_bce_kernel<<<Bz, NV, 0, stream>>>(v, accV, v_target, out_qv, out_loss, NV, 1.0f / (Bz * NV));
}